// GNN_57286273794622
// MI455X (gfx1250) — compile-verified
//
#include <hip/hip_runtime.h>

typedef __attribute__((ext_vector_type(2))) float v2f;
typedef __attribute__((ext_vector_type(8))) float v8f;

#define NODES   50000
#define EDGES   400000
#define DIN     256
#define DHID    512
#define DOUT    128
#define LAYERS  4

// GEMM tiling
#define KB   32          // K-chunk staged in LDS
#define LDA  36          // LDS row stride (floats) for A: 36*i mod 64 = 4*(9i mod 16) -> conflict-free
#define LDB  72          // LDS row stride (floats) for B: 2*72 mod 64 = 16 -> lane halves on disjoint banks

// ---------------------------------------------------------------------------
// Degree / norm precompute
// ---------------------------------------------------------------------------
__global__ __launch_bounds__(256) void deg_init_kernel(int* __restrict__ deg, int n) {
    int i = blockIdx.x * 256 + threadIdx.x;
    if (i < n) deg[i] = 1;                      // self-loop contributes 1
}

__global__ __launch_bounds__(256) void deg_count_kernel(const int* __restrict__ dst,
                                                        int* __restrict__ deg, int e) {
    int i = blockIdx.x * 256 + threadIdx.x;
    if (i < e) atomicAdd(&deg[dst[i]], 1);
}

__global__ __launch_bounds__(256) void dinv_kernel(const int* __restrict__ deg,
                                                   float* __restrict__ dinv, int n) {
    int i = blockIdx.x * 256 + threadIdx.x;
    if (i < n) dinv[i] = rsqrtf((float)deg[i]);
}

// ---------------------------------------------------------------------------
// Async global -> LDS copy of 16 bytes (per lane), tracked by ASYNCcnt.
// ---------------------------------------------------------------------------
__device__ __forceinline__ void async_ld16(unsigned lds_off, const float* gp) {
    unsigned long long ga = (unsigned long long)(uintptr_t)gp;
    asm volatile("global_load_async_to_lds_b128 %0, %1, off"
                 :: "v"(lds_off), "v"(ga) : "memory");
}

__device__ __forceinline__ unsigned lds_u32(const void* p) {
    return (unsigned)(uintptr_t)p;   // LDS aperture: low 32 bits = LDS byte offset
}

// ---------------------------------------------------------------------------
// fp32 WMMA GEMM with async-LDS double buffering:
//   C[M,N] = act(A[M,K] @ B[K,N] (+ bias))
// 256 threads = 8 waves; block tile 128(M) x 64(N); wave tile 64(M) x 16(N).
// K is consumed in KB=32 chunks staged via GLOBAL_LOAD_ASYNC_TO_LDS_B128.
// ---------------------------------------------------------------------------
__global__ __launch_bounds__(256)
void gemm_f32_wmma_kernel(const float* __restrict__ A, const float* __restrict__ B,
                          const float* __restrict__ bias, float* __restrict__ C,
                          int M, int K, int N, int do_relu) {
    __shared__ float lsA[2][128 * LDA];   // 128 x KB tile of A (padded rows)
    __shared__ float lsB[2][KB * LDB];    // KB x 64 tile of B (padded rows)

    const int tid  = threadIdx.x;
    const int wave = tid >> 5;          // 0..7
    const int lane = tid & 31;
    const int half = lane >> 4;         // 0 or 1 (K sub-pair selector)
    const int l16  = lane & 15;
    const int wm   = wave >> 2;         // 0..1  (M direction)
    const int wn   = wave & 3;          // 0..3  (N direction)

    const int rowBase0 = blockIdx.x * 128;
    const int colBase0 = blockIdx.y * 64;
    const int colLocal = wn * 16 + l16;
    const int col      = colBase0 + colLocal;
    const int koff     = 2 * half;      // lanes 0-15 hold K={0,1}, 16-31 K={2,3}

    v8f acc[4];
#pragma unroll
    for (int m = 0; m < 4; ++m)
#pragma unroll
        for (int g = 0; g < 8; ++g) acc[m][g] = 0.0f;

    const int nkb = K / KB;

    // ---- stage chunk kb into buffer buf (all 256 threads, EXEC all-1s) ----
    auto stage = [&](int buf, int kb) {
        const int base_k = kb * KB;
        // A tile: 128 rows x KB cols = 1024 float4
#pragma unroll
        for (int i = 0; i < 4; ++i) {
            int idx = tid + i * 256;
            int r = idx >> 3, c4 = idx & 7;              // 8 float4 per row
            int grow = rowBase0 + r; grow = (grow < M) ? grow : (M - 1);
            async_ld16(lds_u32(&lsA[buf][r * LDA + c4 * 4]),
                       A + (size_t)grow * K + base_k + c4 * 4);
        }
        // B tile: KB rows x 64 cols = 512 float4
#pragma unroll
        for (int i = 0; i < 2; ++i) {
            int idx = tid + i * 256;
            int r = idx >> 4, c4 = idx & 15;             // 16 float4 per row
            async_ld16(lds_u32(&lsB[buf][r * LDB + c4 * 4]),
                       B + (size_t)(base_k + r) * N + colBase0 + c4 * 4);
        }
    };

    stage(0, 0);                                          // prologue prefetch

    for (int kb = 0; kb < nkb; ++kb) {
        const int cur = kb & 1;
        // my async transfers for buffer `cur` complete:
        asm volatile("s_wait_asynccnt 0x0" ::: "memory");
        // everyone's complete, and everyone is done reading buffer cur^1:
        __syncthreads();
        if (kb + 1 < nkb) stage(cur ^ 1, kb + 1);         // overlap next chunk

#pragma unroll
        for (int kk = 0; kk < KB; kk += 4) {
            v2f b;
            b.x = lsB[cur][(kk + koff)     * LDB + colLocal];
            b.y = lsB[cur][(kk + koff + 1) * LDB + colLocal];
#pragma unroll
            for (int m = 0; m < 4; ++m) {
                const float* ap = &lsA[cur][(wm * 64 + m * 16 + l16) * LDA + kk + koff];
                v2f a = *(const v2f*)ap;                  // ds_load_b64, conflict-free
                acc[m] = __builtin_amdgcn_wmma_f32_16x16x4_f32(
                    false, a, false, b, (short)0, acc[m], false, false);
            }
        }
    }

    // Store: C/D layout -> VGPR g holds rows g (lanes 0-15) and g+8 (lanes 16-31)
    const float bv = bias ? bias[col] : 0.0f;
#pragma unroll
    for (int m = 0; m < 4; ++m) {
#pragma unroll
        for (int g = 0; g < 8; ++g) {
            int row = rowBase0 + wm * 64 + m * 16 + g + 8 * half;
            if (row < M) {
                float v = acc[m][g] + bv;
                if (do_relu) v = v > 0.0f ? v : 0.0f;
                C[(size_t)row * N + col] = v;
            }
        }
    }
}

// ---------------------------------------------------------------------------
// acc[n, :] = bias[:]   (layer bias pre-load, scatter accumulates on top)
// ---------------------------------------------------------------------------
__global__ __launch_bounds__(256)
void init_bias_kernel(float* __restrict__ acc, const float* __restrict__ bias,
                      int n_nodes, int d) {
    size_t i = (size_t)blockIdx.x * 256 + threadIdx.x;   // over n_nodes*d/4
    size_t total = (size_t)n_nodes * d / 4;
    if (i >= total) return;
    int f = (int)(i % (d / 4)) * 4;
    float4 bv = make_float4(bias[f], bias[f + 1], bias[f + 2], bias[f + 3]);
    ((float4*)acc)[i] = bv;
}

// ---------------------------------------------------------------------------
// Message scatter: acc[dst] += dinv[src]*dinv[dst] * hw[src]
// One block per message (E real edges then N self-loops); 128 threads x float4.
// ---------------------------------------------------------------------------
__global__ __launch_bounds__(128)
void gcn_scatter_kernel(const float* __restrict__ hw, const int* __restrict__ src,
                        const int* __restrict__ dst, const float* __restrict__ dinv,
                        float* __restrict__ acc, int e, int d) {
    int m = blockIdx.x;
    int s, t;
    if (m < e) { s = src[m]; t = dst[m]; }
    else       { s = t = m - e; }                 // self loop
    float nrm = dinv[s] * dinv[t];
    int f = threadIdx.x * 4;                      // blockDim.x == d/4
    float4 v = *(const float4*)(hw + (size_t)s * d + f);
    float* a = acc + (size_t)t * d + f;
    atomicAdd(a + 0, nrm * v.x);
    atomicAdd(a + 1, nrm * v.y);
    atomicAdd(a + 2, nrm * v.z);
    atomicAdd(a + 3, nrm * v.w);
}

// ---------------------------------------------------------------------------
// Launch
// ---------------------------------------------------------------------------
static inline size_t align256(size_t x) { return (x + 255) & ~(size_t)255; }

extern "C" void kernel_launch(void* const* d_in, const int* in_sizes, int n_in,
                              void* d_out, int out_size, void* d_ws, size_t ws_size,
                              hipStream_t stream) {
    const float* x    = (const float*)d_in[0];                 // [NODES, DIN]
    const int*   eidx = (const int*)d_in[1];                   // [2, EDGES]
    const float* W1   = (const float*)d_in[2];                 // [DIN, DHID]
    const float* b1   = (const float*)d_in[3];                 // [DHID]
    const float* Wc   = (const float*)d_in[4];                 // [LAYERS, DHID, DHID]
    const float* bc   = (const float*)d_in[5];                 // [LAYERS, DHID]
    const float* W2   = (const float*)d_in[6];                 // [DHID, DOUT]
    const float* b2   = (const float*)d_in[7];                 // [DOUT]
    float*       out  = (float*)d_out;                         // [NODES, DOUT]

    const int* src = eidx;                                     // edge_index[0]
    const int* dst = eidx + EDGES;                             // edge_index[1]

    // Workspace carve-up
    char* ws = (char*)d_ws;
    size_t off = 0;
    int*   deg  = (int*)(ws + off);   off += align256((size_t)NODES * sizeof(int));
    float* dinv = (float*)(ws + off); off += align256((size_t)NODES * sizeof(float));
    float* bufA = (float*)(ws + off); off += align256((size_t)NODES * DHID * sizeof(float));
    float* bufB = (float*)(ws + off); off += align256((size_t)NODES * DHID * sizeof(float));
    float* bufC = (float*)(ws + off); off += align256((size_t)NODES * DHID * sizeof(float));
    (void)ws_size; (void)n_in; (void)in_sizes; (void)out_size;

    // 1) degrees & norms (integer atomics -> deterministic)
    deg_init_kernel<<<(NODES + 255) / 256, 256, 0, stream>>>(deg, NODES);
    deg_count_kernel<<<(EDGES + 255) / 256, 256, 0, stream>>>(dst, deg, EDGES);
    dinv_kernel<<<(NODES + 255) / 256, 256, 0, stream>>>(deg, dinv, NODES);

    const int mblk = (NODES + 127) / 128;   // 391

    // 2) dnn1: h = relu(x @ W1 + b1)
    float* h   = bufA;
    float* hw  = bufB;
    float* acc = bufC;
    gemm_f32_wmma_kernel<<<dim3(mblk, DHID / 64), 256, 0, stream>>>(
        x, W1, b1, h, NODES, DIN, DHID, 1);

    // 3) GCN layers
    const int n_msgs = EDGES + NODES;
    for (int l = 0; l < LAYERS; ++l) {
        gemm_f32_wmma_kernel<<<dim3(mblk, DHID / 64), 256, 0, stream>>>(
            h, Wc + (size_t)l * DHID * DHID, nullptr, hw, NODES, DHID, DHID, 0);
        init_bias_kernel<<<(int)(((size_t)NODES * DHID / 4 + 255) / 256), 256, 0, stream>>>(
            acc, bc + (size_t)l * DHID, NODES, DHID);
        gcn_scatter_kernel<<<n_msgs, DHID / 4, 0, stream>>>(
            hw, src, dst, dinv, acc, EDGES, DHID);
        // ping-pong: new h is acc; old h becomes next acc
        float* tmp = h; h = acc; acc = tmp;
    }

    // 4) dnn2: out = h @ W2 + b2
    gemm_f32_wmma_kernel<<<dim3(mblk, DOUT / 64), 256, 0, stream>>>(
        h, W2, b2, out, NODES, DHID, DOUT, 0);
}